// HMM_69612829934002
// MI455X (gfx1250) — compile-verified
//
#include <hip/hip_runtime.h>

typedef __attribute__((ext_vector_type(2))) float v2f;
typedef __attribute__((ext_vector_type(8))) float v8f;

#define N_STATES 35
#define DIM      2496
#define TLEN     512
#define BATCH    64

// ---------------------------------------------------------------------------
// Kernel 1: m_sq[s] = sum_d m[s][d]^2, zero-padded to 48 entries.
// ---------------------------------------------------------------------------
__global__ void hmm_msq_kernel(const float* __restrict__ m, float* __restrict__ msq) {
  int s = threadIdx.x;
  if (s < 48) {
    float acc = 0.f;
    if (s < N_STATES) {
      const float* row = m + (size_t)s * DIM;
      for (int k = 0; k < DIM; ++k) acc = fmaf(row[k], row[k], acc);
    }
    msq[s] = acc;
  }
}

// ---------------------------------------------------------------------------
// Kernel 2: emissions via f32 WMMA.
// emis[row][s] = (|x_row|^2 + |m_s|^2 - 2 * x_row . m_s) / (-500)
// One wave per 16-row tile of the flattened [32768 x 2496] observation matrix.
// N = 35 states padded to 3 tiles of 16 (clamped loads, unstored columns).
// ---------------------------------------------------------------------------
__global__ __launch_bounds__(256) void hmm_emis_kernel(const float* __restrict__ x,
                                                       const float* __restrict__ m,
                                                       const float* __restrict__ msq,
                                                       float* __restrict__ emis) {
  const int wave = threadIdx.x >> 5;
  const int lane = threadIdx.x & 31;
  const int tile = blockIdx.x * 8 + wave;      // 256 blocks * 8 waves = 2048 tiles
  const int half = lane >> 4;                  // 0: K-offset 0, 1: K-offset 2
  const int l15  = lane & 15;
  const int koff = half * 2;

  // A fragment (16x4 f32): lane half selects K in {0,2}; 2 consecutive floats.
  const int rowA = tile * 16 + l15;
  const float* xrow  = x + (size_t)rowA * DIM + koff;
  // B fragments (4x16 f32): lane = column N, lane half selects K in {0,2}.
  const float* mrow0 = m + (size_t)l15 * DIM + koff;                 // s = l15
  const float* mrow1 = m + (size_t)(16 + l15) * DIM + koff;          // s = 16+l15
  const int n2 = 32 + l15;
  const float* mrow2 = m + (size_t)(n2 < N_STATES ? n2 : N_STATES - 1) * DIM + koff;

  v8f c0 = {}; v8f c1 = {}; v8f c2 = {};
  float xsq = 0.f;

  for (int k = 0; k < DIM; k += 4) {
    v2f a  = *(const v2f*)(xrow  + k);
    v2f b0 = *(const v2f*)(mrow0 + k);
    v2f b1 = *(const v2f*)(mrow1 + k);
    v2f b2 = *(const v2f*)(mrow2 + k);
    xsq = fmaf(a[0], a[0], xsq);
    xsq = fmaf(a[1], a[1], xsq);
    c0 = __builtin_amdgcn_wmma_f32_16x16x4_f32(false, a, false, b0, (short)0, c0, false, false);
    c1 = __builtin_amdgcn_wmma_f32_16x16x4_f32(false, a, false, b1, (short)0, c1, false, false);
    c2 = __builtin_amdgcn_wmma_f32_16x16x4_f32(false, a, false, b2, (short)0, c2, false, false);
  }
  // Row (l15) |x|^2 = lane part (K%4 in {0,1}) + partner part (K%4 in {2,3}).
  xsq += __shfl_xor(xsq, 16, 32);

  const float ms0 = msq[l15];
  const float ms1 = msq[16 + l15];
  const float ms2 = msq[32 + l15];             // zero-padded for s >= 35
  const float scale = -1.0f / 500.0f;

  #pragma unroll
  for (int v = 0; v < 8; ++v) {
    const int M = v + half * 8;                // C layout: VGPR v, lane half -> row
    const float xs = __shfl(xsq, M, 32);       // |x|^2 of output row M lives at lane M
    float* erow = emis + (size_t)(tile * 16 + M) * N_STATES;
    erow[l15]      = (xs + ms0 - 2.f * c0[v]) * scale;
    erow[16 + l15] = (xs + ms1 - 2.f * c1[v]) * scale;
    if (n2 < N_STATES)
      erow[n2]     = (xs + ms2 - 2.f * c2[v]) * scale;
  }
}

// ---------------------------------------------------------------------------
// Kernel 3: forward recursion, one block per batch.
// Keep p[i] = exp(alpha[i] - C); step is a plain dot with the (normalized)
// transition probability matrix -> 1 exp + 1 log per state per step.
// ---------------------------------------------------------------------------
__global__ __launch_bounds__(64) void hmm_forward_kernel(const float* __restrict__ emis,
                                                         float* __restrict__ out) {
  const int b = blockIdx.x;
  const int j = threadIdx.x;
  __shared__ float p[N_STATES + 1];
  __shared__ float nbuf[N_STATES + 1];

  // trans[i][j] = base + 0.1*(j==i) + 0.8*(j==(i+1)%35) + 0.08*(j==(i+2)%35)
  const float base = 0.02f / 35.0f;
  float rs = 0.f;
  for (int t = 0; t < N_STATES; ++t) rs += base;
  rs += 0.1f; rs += 0.8f; rs += 0.08f;
  const float inv_rs = 1.0f / rs;

  float trj[N_STATES];                          // column j of normalized trans
  #pragma unroll
  for (int i = 0; i < N_STATES; ++i) {
    float v = base;
    if (j == i) v += 0.1f;
    if (j == (i + 1) % N_STATES) v += 0.8f;
    if (j == (i + 2) % N_STATES) v += 0.08f;
    trj[i] = v * inv_rs;
  }

  const float* e = emis + (size_t)b * TLEN * N_STATES;

  // alpha0 = e[0] + prior
  if (j < N_STATES) nbuf[j] = e[j] + (j == 0 ? 10.0f : -10.0f);
  __syncthreads();

  float c = -3.4e38f;
  for (int i = 0; i < N_STATES; ++i) c = fmaxf(c, nbuf[i]);   // all threads agree
  float C = c;
  if (j < N_STATES) p[j] = __expf(nbuf[j] - c);
  __syncthreads();

  for (int t = 1; t < TLEN; ++t) {
    const float* et = e + (size_t)t * N_STATES;
    float nj = 0.f;
    if (j < N_STATES) {
      float s = 0.f;
      #pragma unroll
      for (int i = 0; i < N_STATES; ++i) s = fmaf(trj[i], p[i], s);
      nj = et[j] + __logf(s);                   // alpha_new[j] - C
      nbuf[j] = nj;
    }
    __syncthreads();
    c = -3.4e38f;
    for (int i = 0; i < N_STATES; ++i) c = fmaxf(c, nbuf[i]);
    C += c;
    if (j < N_STATES) p[j] = __expf(nj - c);
    __syncthreads();
  }

  if (j == 0) {
    float s = 0.f;
    for (int i = 0; i < N_STATES; ++i) s += p[i];
    out[b] = C + __logf(s);                     // logsumexp over states
  }
}

// ---------------------------------------------------------------------------
extern "C" void kernel_launch(void* const* d_in, const int* in_sizes, int n_in,
                              void* d_out, int out_size, void* d_ws, size_t ws_size,
                              hipStream_t stream) {
  const float* x = (const float*)d_in[0];   // [64,512,16,13,12] f32
  const float* m = (const float*)d_in[1];   // [35,16,13,12] f32
  float* msq  = (float*)d_ws;               // 48 floats (padded to 64)
  float* emis = msq + 64;                   // [32768, 35] f32 (~4.6 MB)

  hmm_msq_kernel<<<1, 64, 0, stream>>>(m, msq);
  hmm_emis_kernel<<<256, 256, 0, stream>>>(x, m, msq, emis);
  hmm_forward_kernel<<<BATCH, 64, 0, stream>>>(emis, (float*)d_out);
}